// SparseCrossAttention_996432412692
// MI455X (gfx1250) — compile-verified
//
#include <hip/hip_runtime.h>

// ---------------------------------------------------------------------------
// SparseCrossAttention on MI455X (gfx1250, wave32, WMMA).
// Dense interpretation: per (b,h) attention over P with a per-head score-mix
// MLP. All matmuls use V_WMMA_F32_16X16X4_F32 (full f32 fidelity): at ~1 GFLOP
// and ~6 MB of L2-resident data, the problem is latency-bound, not rate-bound.
// v2: fused QKV (3 launches total), 16x64 wave tiles with 4 independent WMMA
// accumulator chains, K-split attn@V across all 8 waves, cost prefetch.
// ---------------------------------------------------------------------------

typedef __attribute__((ext_vector_type(2))) float v2f;
typedef __attribute__((ext_vector_type(8))) float v8f;

constexpr int kBS = 4;
constexpr int kS  = 256;
constexpr int kP  = 256;
constexpr int kD  = 256;
constexpr int kH  = 8;
constexpr int kHD = 32;
constexpr int kMSH = 16;

__device__ __forceinline__ v8f wmma_f32x4(v2f a, v2f b, v8f c) {
  // V_WMMA_F32_16X16X4_F32: D = A(16x4) * B(4x16) + C(16x16)
  return __builtin_amdgcn_wmma_f32_16x16x4_f32(
      /*neg_a=*/false, a, /*neg_b=*/false, b,
      /*c_mod=*/(short)0, c, /*reuse_a=*/false, /*reuse_b=*/false);
}

// One wave computes a 16x64 tile of Y = X @ W^T + bias (W row-major [256,256],
// so B[k][n] = W[n][k]). One A-fragment feeds 4 B-fragments -> 4 independent
// accumulator chains for the XDL pipe.
__device__ __forceinline__ void gemm16x64(const float* __restrict__ X,
                                          const float* __restrict__ W,
                                          const float* __restrict__ bias,
                                          float* __restrict__ Y,
                                          int mt, int ng, int lane) {
  const int half = lane >> 4;   // 0: K=k0,k0+1   1: K=k0+2,k0+3
  const int l16  = lane & 15;
  const float* arow = X + (size_t)(mt * 16 + l16) * kD;
  const float* w0 = W + (size_t)(ng * 64 +  0 + l16) * kD;
  const float* w1 = W + (size_t)(ng * 64 + 16 + l16) * kD;
  const float* w2 = W + (size_t)(ng * 64 + 32 + l16) * kD;
  const float* w3 = W + (size_t)(ng * 64 + 48 + l16) * kD;

  v8f acc0 = {}, acc1 = {}, acc2 = {}, acc3 = {};
#pragma unroll 2
  for (int k0 = 0; k0 < kD; k0 += 4) {
    const int ka = k0 + half * 2;
    v2f a  = {arow[ka], arow[ka + 1]};
    v2f b0 = {w0[ka], w0[ka + 1]};
    v2f b1 = {w1[ka], w1[ka + 1]};
    v2f b2 = {w2[ka], w2[ka + 1]};
    v2f b3 = {w3[ka], w3[ka + 1]};
    acc0 = wmma_f32x4(a, b0, acc0);
    acc1 = wmma_f32x4(a, b1, acc1);
    acc2 = wmma_f32x4(a, b2, acc2);
    acc3 = wmma_f32x4(a, b3, acc3);
  }

  v8f accs[4] = {acc0, acc1, acc2, acc3};
#pragma unroll
  for (int j = 0; j < 4; ++j) {
    const int n = ng * 64 + j * 16 + l16;
    const float bn = bias[n];
#pragma unroll
    for (int r = 0; r < 8; ++r) {
      const int mm = mt * 16 + half * 8 + r;
      Y[(size_t)mm * kD + n] = accs[j][r] + bn;
    }
  }
}

// Fused Q/K/V projections: 3 GEMMs x 256 wave-tiles (64 mt x 4 ngroups).
// 768 wave-tiles / 8 waves per block -> 96 blocks.
__global__ __launch_bounds__(256)
void sca_qkv(const float* __restrict__ row_emb, const float* __restrict__ col_emb,
             const float* __restrict__ Wq, const float* __restrict__ bq,
             const float* __restrict__ Wk, const float* __restrict__ bk,
             const float* __restrict__ Wv, const float* __restrict__ bv,
             float* __restrict__ Qw, float* __restrict__ Kw,
             float* __restrict__ Vw) {
  const int lane = threadIdx.x & 31;
  const int wave = threadIdx.x >> 5;
  const int gw = blockIdx.x * 8 + wave;     // 0..767
  const int which = gw >> 8;                // 0=Q, 1=K, 2=V (wave-uniform)
  const int t  = gw & 255;
  const int mt = t >> 2;
  const int ng = t & 3;
  const float* X    = (which == 0) ? row_emb : col_emb;
  const float* W    = (which == 0) ? Wq : (which == 1) ? Wk : Wv;
  const float* bias = (which == 0) ? bq : (which == 1) ? bk : bv;
  float*       Y    = (which == 0) ? Qw : (which == 1) ? Kw : Vw;
  gemm16x64(X, W, bias, Y, mt, ng, lane);
}

// Output projection: out = heads @ Wo^T + bo. 256 wave-tiles -> 32 blocks.
__global__ __launch_bounds__(256)
void sca_out(const float* __restrict__ Hm, const float* __restrict__ Wo,
             const float* __restrict__ bo, float* __restrict__ out) {
  const int lane = threadIdx.x & 31;
  const int wave = threadIdx.x >> 5;
  const int gw = blockIdx.x * 8 + wave;     // 0..255
  gemm16x64(Hm, Wo, bo, out, gw >> 2, gw & 3, lane);
}

// One block = (b, h, 32-row S slab). WMMA scores -> per-head MLP mix with
// cost -> row softmax over P in LDS -> WMMA attn@V (K-split over wave pairs).
#define LSTR 260   // 32 x 260 floats; stride%64==4 -> 16 rows conflict-free
__global__ __launch_bounds__(256)
void sca_attn(const float* __restrict__ Q, const float* __restrict__ Km,
              const float* __restrict__ Vm, const float* __restrict__ cost,
              const float* __restrict__ mW1, const float* __restrict__ mb1,
              const float* __restrict__ mW2, const float* __restrict__ mb2,
              float* __restrict__ heads) {
  __shared__ float sm[32 * LSTR];
  __shared__ float part[2][32][33];

  const int tid  = threadIdx.x;
  const int lane = tid & 31;
  const int wave = tid >> 5;
  const int half = lane >> 4;
  const int l16  = lane & 15;

  const int blk = blockIdx.x;
  const int b  = blk >> 6;          // 64 blocks per batch (H * S/32)
  const int h  = (blk >> 3) & 7;
  const int s0 = (blk & 7) * 32;

  // Prefetch this block's cost slab (32 KB) so phase 2 hits cache.
  // Lowers to global_prefetch_b8; 256 threads x 128 B covers the slab.
  {
    const char* cpf = (const char*)(cost + (size_t)(b * kS + s0) * kP);
    __builtin_prefetch(cpf + (size_t)tid * 128, 0, 1);
  }

  // ---- Phase 1: raw logits[32,256] = Q_h(32x32) @ K_h^T(32x256) ------------
  // Wave w: mt = w&1, nt = (w>>1)*4 .. +3  (1 A-frag feeds 4 B-frags).
  {
    const int mt  = wave & 1;
    const int ntb = (wave >> 1) * 4;
    const float* qrow = Q + (size_t)(b * kS + s0 + mt * 16 + l16) * kD + h * kHD;
    const float* k0r = Km + (size_t)(b * kP + (ntb + 0) * 16 + l16) * kD + h * kHD;
    const float* k1r = Km + (size_t)(b * kP + (ntb + 1) * 16 + l16) * kD + h * kHD;
    const float* k2r = Km + (size_t)(b * kP + (ntb + 2) * 16 + l16) * kD + h * kHD;
    const float* k3r = Km + (size_t)(b * kP + (ntb + 3) * 16 + l16) * kD + h * kHD;
    v8f a0 = {}, a1 = {}, a2 = {}, a3 = {};
#pragma unroll
    for (int k0 = 0; k0 < kHD; k0 += 4) {
      const int ka = k0 + half * 2;
      v2f a  = {qrow[ka], qrow[ka + 1]};
      v2f b0 = {k0r[ka], k0r[ka + 1]};
      v2f b1 = {k1r[ka], k1r[ka + 1]};
      v2f b2 = {k2r[ka], k2r[ka + 1]};
      v2f b3 = {k3r[ka], k3r[ka + 1]};
      a0 = wmma_f32x4(a, b0, a0);
      a1 = wmma_f32x4(a, b1, a1);
      a2 = wmma_f32x4(a, b2, a2);
      a3 = wmma_f32x4(a, b3, a3);
    }
    v8f accs[4] = {a0, a1, a2, a3};
#pragma unroll
    for (int j = 0; j < 4; ++j)
#pragma unroll
      for (int r = 0; r < 8; ++r)
        sm[(mt * 16 + half * 8 + r) * LSTR + (ntb + j) * 16 + l16] = accs[j][r];
  }
  __syncthreads();

  // ---- Phase 2: per-head MLP([score/sqrt(32), cost]) -----------------------
  float W1a[kMSH], W1b[kMSH], B1[kMSH], W2[kMSH];
#pragma unroll
  for (int j = 0; j < kMSH; ++j) {
    W1a[j] = mW1[h * 2 * kMSH + j];          // weight on dot score
    W1b[j] = mW1[h * 2 * kMSH + kMSH + j];   // weight on edge cost
    B1[j]  = mb1[h * kMSH + j];
    W2[j]  = mW2[h * kMSH + j];
  }
  const float B2 = mb2[h];
  const float iscale = 0.17677669529663687f;  // 1/sqrt(32)

  for (int idx = tid; idx < 32 * kP; idx += 256) {
    const int row = idx >> 8, col = idx & 255;
    const float l = sm[row * LSTR + col] * iscale;
    const float c = cost[(size_t)(b * kS + s0 + row) * kP + col];
    float o = B2;
#pragma unroll
    for (int j = 0; j < kMSH; ++j) {
      const float hid = fmaf(l, W1a[j], fmaf(c, W1b[j], B1[j]));
      o = fmaf(fmaxf(hid, 0.0f), W2[j], o);
    }
    sm[row * LSTR + col] = o;
  }
  __syncthreads();

  // ---- Phase 3: row softmax over P=256 (4 rows per wave, shfl reductions) --
  for (int r = wave * 4; r < wave * 4 + 4; ++r) {
    float* rowp = sm + r * LSTR;
    float vals[8];
    float vmax = -3.4e38f;
#pragma unroll
    for (int i = 0; i < 8; ++i) {
      vals[i] = rowp[lane + i * 32];
      vmax = fmaxf(vmax, vals[i]);
    }
#pragma unroll
    for (int m = 16; m; m >>= 1) vmax = fmaxf(vmax, __shfl_xor(vmax, m, 32));
    float vsum = 0.0f;
#pragma unroll
    for (int i = 0; i < 8; ++i) {
      vals[i] = __expf(vals[i] - vmax);
      vsum += vals[i];
    }
#pragma unroll
    for (int m = 16; m; m >>= 1) vsum += __shfl_xor(vsum, m, 32);
    const float inv = 1.0f / vsum;
#pragma unroll
    for (int i = 0; i < 8; ++i) rowp[lane + i * 32] = vals[i] * inv;
  }
  __syncthreads();

  // ---- Phase 4: heads[32,32] = attn[32,256] @ V_h[256,32] ------------------
  // All 8 waves: tile = wave>>1 (2 mt x 2 nt), kh = wave&1 splits P in half.
  {
    const int tile = wave >> 1;
    const int kh = wave & 1;
    const int mt = tile >> 1, nt = tile & 1;
    const int n = nt * 16 + l16;
    const float* vb = Vm + (size_t)(b * kP + kh * 128) * kD + h * kHD + n;
    const float* arow = sm + (mt * 16 + l16) * LSTR + kh * 128;
    v8f acc = {};
#pragma unroll 4
    for (int k0 = 0; k0 < 128; k0 += 4) {
      const int ka = k0 + half * 2;
      v2f a  = {arow[ka], arow[ka + 1]};
      v2f vv = {vb[(size_t)ka * kD], vb[(size_t)(ka + 1) * kD]};
      acc = wmma_f32x4(a, vv, acc);
    }
#pragma unroll
    for (int r = 0; r < 8; ++r)
      part[kh][mt * 16 + half * 8 + r][n] = acc[r];
  }
  __syncthreads();

  // Reduce the two K-halves and write heads[b, s0+row, h*32+col].
  for (int idx = tid; idx < 32 * 32; idx += 256) {
    const int row = idx >> 5, col = idx & 31;
    heads[(size_t)(b * kS + s0 + row) * kD + h * kHD + col] =
        part[0][row][col] + part[1][row][col];
  }
}

extern "C" void kernel_launch(void* const* d_in, const int* in_sizes, int n_in,
                              void* d_out, int out_size, void* d_ws, size_t ws_size,
                              hipStream_t stream) {
  const float* row_emb = (const float*)d_in[0];
  const float* col_emb = (const float*)d_in[1];
  const float* cost    = (const float*)d_in[2];
  const float* Wq = (const float*)d_in[3];
  const float* bq = (const float*)d_in[4];
  const float* Wk = (const float*)d_in[5];
  const float* bk = (const float*)d_in[6];
  const float* Wv = (const float*)d_in[7];
  const float* bv = (const float*)d_in[8];
  const float* Wo = (const float*)d_in[9];
  const float* bo = (const float*)d_in[10];
  const float* mW1 = (const float*)d_in[11];
  const float* mb1 = (const float*)d_in[12];
  const float* mW2 = (const float*)d_in[13];
  const float* mb2 = (const float*)d_in[14];

  const size_t plane = (size_t)kBS * kS * kD;   // 1024*256 floats = 1 MB
  float* Qw = (float*)d_ws;
  float* Kw = Qw + plane;
  float* Vw = Kw + plane;
  float* Hw = Vw + plane;

  sca_qkv<<<96, 256, 0, stream>>>(row_emb, col_emb, Wq, bq, Wk, bk, Wv, bv,
                                  Qw, Kw, Vw);
  sca_attn<<<kBS * kH * (kS / 32), 256, 0, stream>>>(Qw, Kw, Vw, cost,
                                                     mW1, mb1, mW2, mb2, Hw);
  sca_out<<<32, 256, 0, stream>>>(Hw, Wo, bo, (float*)d_out);
}